// FCOS_56032143343798
// MI455X (gfx1250) — compile-verified
//
#include <hip/hip_runtime.h>
#include <hip/hip_bf16.h>
#include <stdint.h>

#define NB          8192
#define NW          (NB / 32)          /* 256 mask words per row */
#define NWIN        (NB / 32)          /* 256 windows of 32 rows  */
#define WPB         4                  /* column words per mask block */
#define IOU_THRESH  0.5f

typedef float v2f  __attribute__((ext_vector_type(2)));
typedef float v8f  __attribute__((ext_vector_type(8)));
typedef unsigned int u32x4 __attribute__((ext_vector_type(4)));
typedef int   i32x4 __attribute__((ext_vector_type(4)));
typedef int   i32x8 __attribute__((ext_vector_type(8)));

// Monotonic float -> uint mapping (order-preserving), for atomicMax on floats.
__device__ __forceinline__ unsigned fmono(float f) {
    unsigned u = __float_as_uint(f);
    return (u & 0x80000000u) ? ~u : (u | 0x80000000u);
}
__device__ __forceinline__ float fmono_inv(unsigned u) {
    unsigned b = (u & 0x80000000u) ? (u ^ 0x80000000u) : ~u;
    return __uint_as_float(b);
}

// ---------------------------------------------------------------------------
// Tensor Data Mover: load a 2-D tile (tile0 x tile1 elements of 4 B) from
// global (row stride stride0 elements) into LDS at byte address ldsaddr.
// tile1 == 0 -> 1-D tile. Issued once per executing wave; tracked by TENSORcnt.
// ---------------------------------------------------------------------------
__device__ __forceinline__ void tdm_load_2d(const void* gsrc, unsigned ldsaddr,
                                            unsigned tile0, unsigned tile1,
                                            unsigned tdim0, unsigned tdim1,
                                            unsigned stride0)
{
    unsigned long long ga = (unsigned long long)(uintptr_t)gsrc;
    u32x4 g0;
    g0[0] = 1u;                                        // count=1, user D#
    g0[1] = ldsaddr;                                   // lds_addr (bytes)
    g0[2] = (unsigned)(ga & 0xFFFFFFFFull);            // global_addr[31:0]
    g0[3] = (unsigned)((ga >> 32) & 0x1FFFFFFull)      // global_addr[56:32]
          | 0x80000000u;                               // type = 2 ("image")
    i32x8 g1;
    g1[0] = (int)(2u << 16);                           // data_size = 4 B
    g1[1] = (int)((tdim0 & 0xFFFFu) << 16);            // tensor_dim0[15:0]
    g1[2] = (int)((tdim0 >> 16) | ((tdim1 & 0xFFFFu) << 16));
    g1[3] = (int)((tdim1 >> 16) | ((tile0 & 0xFFFFu) << 16)); // tile_dim0
    g1[4] = (int)(tile1 & 0xFFFFu);                    // tile_dim1 (0=1-D)
    g1[5] = (int)stride0;                              // tensor_dim0_stride
    g1[6] = 0;
    g1[7] = 0;
    i32x4 z4 = {};
#if defined(__clang_major__) && (__clang_major__ >= 23)
    i32x8 z8 = {};
    __builtin_amdgcn_tensor_load_to_lds(g0, g1, z4, z4, z8, 0);
#else
    __builtin_amdgcn_tensor_load_to_lds(g0, g1, z4, z4, 0);
#endif
}

__global__ void init_kernel(unsigned* maxMono) { *maxMono = 0u; }

// ---------------------------------------------------------------------------
// Box decode via V_WMMA_F32_16X16X4_F32:
//   D[16x16] = clip(deltas)[16x4] x B[4x16],  B = diag(-s,-s,+s,+s) padded
//   box[m][n] = D[m][n] + (n even ? xc[m] : yc[m]),  n < 4
// One wave decodes 16 boxes. Also reduces max coordinate (monotonic atomicMax).
// ---------------------------------------------------------------------------
__global__ __launch_bounds__(128) void decode_kernel(
    const float* __restrict__ deltas, const float* __restrict__ loc,
    const int* __restrict__ stride_p, float* __restrict__ boxes,
    unsigned* __restrict__ maxMono)
{
    const float s = (float)(*stride_p);
    const int lane  = threadIdx.x & 31;
    const int wave  = threadIdx.x >> 5;
    const int wbase = (blockIdx.x * (blockDim.x >> 5) + wave) * 16;
    const int m     = lane & 15;
    const bool hi   = lane >= 16;
    const int box   = wbase + m;

    // A-matrix 16x4 f32: VGPR0 = K0 (lanes 0-15) / K2 (lanes 16-31), VGPR1 = K1/K3
    v2f a;
    a.x = fmaxf(deltas[box * 4 + (hi ? 2 : 0)], 0.0f);
    a.y = fmaxf(deltas[box * 4 + (hi ? 3 : 1)], 0.0f);

    // B-matrix 4x16: row k striped across lanes; B[k][n] = (n==k) ? sgn_k*s : 0
    v2f b;
    b.x = (!hi && m == 0) ? -s : ((hi && m == 2) ? s : 0.0f);
    b.y = (!hi && m == 1) ? -s : ((hi && m == 3) ? s : 0.0f);

    v8f c = {};
    v8f d = __builtin_amdgcn_wmma_f32_16x16x4_f32(false, a, false, b,
                                                  (short)0, c, false, false);

    // D layout: VGPR r -> M = r + (hi?8:0), N = lane%16. Only N<4 is the box.
    if (m < 4) {
        unsigned mymax = 0u;
#pragma unroll
        for (int r = 0; r < 8; ++r) {
            int row   = r + (hi ? 8 : 0);
            float ctr = loc[(wbase + row) * 2 + (m & 1)];
            float v   = d[r] + ctr;
            boxes[(wbase + row) * 4 + m] = v;
            unsigned u = fmono(v);
            mymax = mymax > u ? mymax : u;
        }
        atomicMax(maxMono, mymax);
    }
}

// ---------------------------------------------------------------------------
// Bitonic sort (descending by score, ties by ascending index) — one WG,
// 8192 packed u64 keys in 64 KB LDS (WGP has 320 KB).
// ---------------------------------------------------------------------------
__global__ __launch_bounds__(1024) void sort_kernel(
    const float* __restrict__ scores, int* __restrict__ order)
{
    __shared__ unsigned long long key[NB];
    for (int t = threadIdx.x; t < NB; t += 1024) {
        key[t] = ((unsigned long long)fmono(scores[t]) << 32)
               | (unsigned)(NB - 1 - t);          // tie: smaller idx wins
    }
    __syncthreads();
    for (int k = 2; k <= NB; k <<= 1) {
        for (int j = k >> 1; j > 0; j >>= 1) {
            for (int t = threadIdx.x; t < NB; t += 1024) {
                int ixj = t ^ j;
                if (ixj > t) {
                    unsigned long long A = key[t], B = key[ixj];
                    bool descSeg = (t & k) == 0;
                    if (descSeg ? (A < B) : (A > B)) { key[t] = B; key[ixj] = A; }
                }
            }
            __syncthreads();
        }
    }
    for (int t = threadIdx.x; t < NB; t += 1024)
        order[t] = NB - 1 - (int)(key[t] & 0xFFFFFFFFull);
}

// ---------------------------------------------------------------------------
// Gather class-offset boxes in sorted order into SoA [5][NB]: x1,y1,x2,y2,area
// ---------------------------------------------------------------------------
__global__ void gather_kernel(
    const float* __restrict__ boxes, const int* __restrict__ cls,
    const int* __restrict__ order, const unsigned* __restrict__ maxMono,
    float* __restrict__ sorted)
{
    int t = blockIdx.x * blockDim.x + threadIdx.x;
    if (t >= NB) return;
    float maxc = fmono_inv(*maxMono);
    int i = order[t];
    float off = (float)cls[i] * (maxc + 1.0f);
    float x1 = boxes[i * 4 + 0] + off, y1 = boxes[i * 4 + 1] + off;
    float x2 = boxes[i * 4 + 2] + off, y2 = boxes[i * 4 + 3] + off;
    sorted[0 * NB + t] = x1;
    sorted[1 * NB + t] = y1;
    sorted[2 * NB + t] = x2;
    sorted[3 * NB + t] = y2;
    sorted[4 * NB + t] = (x2 - x1) * (y2 - y1);
}

// ---------------------------------------------------------------------------
// Suppression-bit matrix. Block = (32,8) = 8 wave32, covering 8 rows x
// (WPB*32) columns. Column tile (5 x 128 floats of the SoA) staged into LDS
// with one TDM 2-D tile load; each wave emits WPB mask words via ballot,
// reusing its row box across the 4 column words.
// ---------------------------------------------------------------------------
__global__ __launch_bounds__(256) void mask_kernel(
    const float* __restrict__ sorted, unsigned* __restrict__ mask)
{
    __shared__ float col[5 * 32 * WPB];
    const int tx = threadIdx.x, ty = threadIdx.y;
    const int cw0 = blockIdx.x * WPB;        // first column word of this block
    const int r   = blockIdx.y * 8 + ty;     // row (sorted order)

    if (ty == 0) {  // one wave issues the TDM load; EXEC ignored by TDM
        tdm_load_2d(sorted + cw0 * 32,
                    (unsigned)(uintptr_t)(void*)col,
                    /*tile0=*/32 * WPB, /*tile1=*/5,
                    /*tdim0=*/32 * WPB, /*tdim1=*/5, /*stride0=*/NB);
        __builtin_amdgcn_s_wait_tensorcnt(0);
    }
    __syncthreads();

    const float rx1 = sorted[0 * NB + r], ry1 = sorted[1 * NB + r];
    const float rx2 = sorted[2 * NB + r], ry2 = sorted[3 * NB + r];
    const float ra  = sorted[4 * NB + r];

#pragma unroll
    for (int w = 0; w < WPB; ++w) {
        int t = w * 32 + tx;
        float cx1 = col[0 * 32 * WPB + t], cy1 = col[1 * 32 * WPB + t];
        float cx2 = col[2 * 32 * WPB + t], cy2 = col[3 * 32 * WPB + t];
        float ca  = col[4 * 32 * WPB + t];

        float xx1 = fmaxf(rx1, cx1), yy1 = fmaxf(ry1, cy1);
        float xx2 = fminf(rx2, cx2), yy2 = fminf(ry2, cy2);
        float inter = fmaxf(xx2 - xx1, 0.0f) * fmaxf(yy2 - yy1, 0.0f);

        int c = (cw0 + w) * 32 + tx;
        // iou > thr  <=>  inter*(1+thr) > thr*(ra+ca)   (no divide)
        bool bit = (c > r) && (inter * (1.0f + IOU_THRESH) > IOU_THRESH * (ra + ca));

        unsigned word = __builtin_amdgcn_ballot_w32(bit);
        if (tx == 0) mask[r * NW + cw0 + w] = word;
    }
}

// ---------------------------------------------------------------------------
// Suppression scan, one WG of 256 threads. The 8 MB mask matrix is streamed
// through LDS in 32-row (32 KB, contiguous) windows by the TDM, double-
// buffered with s_wait_tensorcnt pipelining. Rows 32w..32w+31 all have their
// "removed" bits inside word w, so thread w resolves the whole group's keep
// bits serially from LDS, broadcasts a 32-bit keep mask, and all threads OR
// the kept rows' words in parallel (3 barriers per window instead of 32).
// ---------------------------------------------------------------------------
__global__ __launch_bounds__(256) void nms_reduce_kernel(
    const unsigned* __restrict__ mask, const int* __restrict__ order,
    float* __restrict__ keep_out)
{
    __shared__ unsigned win[2][32 * NW];   // 2 x 32 KB ping-pong
    __shared__ unsigned removed[NW];
    __shared__ unsigned sh_keep;
    const int tid = threadIdx.x;

    removed[tid] = 0u;
    if (tid < 32) {  // wave0: prime the pipeline with window 0
        tdm_load_2d(mask, (unsigned)(uintptr_t)(void*)&win[0][0],
                    /*tile0=*/32 * NW, /*tile1=*/0,
                    /*tdim0=*/32 * NW, /*tdim1=*/1, /*stride0=*/32 * NW);
    }
    __syncthreads();

    for (int w = 0; w < NWIN; ++w) {
        const int buf = w & 1;
        if (tid < 32) {  // wave0: issue next window, wait for current
            if (w + 1 < NWIN) {
                tdm_load_2d(mask + (unsigned)(w + 1) * 32 * NW,
                            (unsigned)(uintptr_t)(void*)&win[buf ^ 1][0],
                            32 * NW, 0, 32 * NW, 1, 32 * NW);
                __builtin_amdgcn_s_wait_tensorcnt(1);   // window w complete
            } else {
                __builtin_amdgcn_s_wait_tensorcnt(0);
            }
        }
        __syncthreads();                    // window w visible to all

        if (tid == w) {  // resolve intra-group dependencies (all in word w)
            unsigned rm = removed[w];
            unsigned keepm = 0u;
            for (int j = 0; j < 32; ++j) {
                if (!((rm >> j) & 1u)) {
                    keepm |= (1u << j);
                    rm |= win[buf][j * NW + w];
                }
            }
            removed[w] = rm;
            sh_keep = keepm;
        }
        __syncthreads();

        unsigned keepm = sh_keep;
        unsigned acc = removed[tid];
        for (int j = 0; j < 32; ++j)
            if ((keepm >> j) & 1u) acc |= win[buf][j * NW + tid];
        removed[tid] = acc;
        __syncthreads();                    // before next group's leader reads
    }

    for (int t = tid; t < NB; t += 256) {
        bool kp = !((removed[t >> 5] >> (t & 31)) & 1u);
        keep_out[order[t]] = kp ? 1.0f : 0.0f;
    }
}

extern "C" void kernel_launch(void* const* d_in, const int* in_sizes, int n_in,
                              void* d_out, int out_size, void* d_ws, size_t ws_size,
                              hipStream_t stream) {
    const float* deltas  = (const float*)d_in[0];
    const float* loc     = (const float*)d_in[1];
    const float* scores  = (const float*)d_in[2];
    const int*   cls     = (const int*)d_in[3];
    const int*   stridep = (const int*)d_in[4];

    float* out_boxes = (float*)d_out;          // [NB][4]
    float* out_keep  = out_boxes + NB * 4;     // [NB] (bool as 0.0/1.0)

    char* ws = (char*)d_ws;
    unsigned* maxMono = (unsigned*)ws;                 // 4 B
    int*      order   = (int*)(ws + 256);              // 32 KB
    float*    sorted  = (float*)(ws + 33280);          // 5*NB floats = 160 KB
    unsigned* mask    = (unsigned*)(ws + 197120);      // NB*NW u32 = 8 MB

    init_kernel<<<1, 1, 0, stream>>>(maxMono);
    decode_kernel<<<NB / 64, 128, 0, stream>>>(deltas, loc, stridep, out_boxes, maxMono);
    sort_kernel<<<1, 1024, 0, stream>>>(scores, order);
    gather_kernel<<<(NB + 255) / 256, 256, 0, stream>>>(out_boxes, cls, order, maxMono, sorted);
    mask_kernel<<<dim3(NW / WPB, NB / 8), dim3(32, 8), 0, stream>>>(sorted, mask);
    nms_reduce_kernel<<<1, 256, 0, stream>>>(mask, order, out_keep);
}